// BehaviorRegression_22196390986434
// MI455X (gfx1250) — compile-verified
//
#include <hip/hip_runtime.h>

typedef __attribute__((ext_vector_type(2))) float v2f;
typedef __attribute__((ext_vector_type(8))) float v8f;

// ---------------------------------------------------------------------------
// Pass 1: proj[token][d] = dot(feat[token][:], W[d][:])
// One wave = 16 tokens via V_WMMA_F32_16X16X4_F32, K-loop over H in steps of
// 4 (two interleaved accumulators -> 8 K per trip). A tile (16x4 f32) streams
// from HBM (the only real traffic: 128 MiB ~= 5.5 us at 23.3 TB/s). B tile
// (4x16) holds W in columns 0..D-1; columns D..15 read a dedicated zero row
// in LDS so no per-iteration masking VALU is needed.
// ---------------------------------------------------------------------------
__global__ __launch_bounds__(256) void br_proj_wmma_kernel(
    const float* __restrict__ feat, const float* __restrict__ W,
    float* __restrict__ proj, int H, int D)
{
  // rows 0..D-1 = W, row D = zeros; stride H+4 floats to decouple LDS banks
  __shared__ float sW[3 * (1024 + 4)];
  const int HP  = H + 4;
  const int tid = threadIdx.x;
  for (int i = tid; i < (D + 1) * HP; i += blockDim.x) {
    const int r = i / HP, k = i - r * HP;
    sW[i] = (r < D && k < H) ? W[r * H + k] : 0.0f;
  }
  __syncthreads();

  const int lane = tid & 31;
  const int wave = tid >> 5;
  const int half = lane >> 4;                 // 0 -> K {0,1}, 1 -> K {2,3}
  const int mn   = lane & 15;                 // A: row M ; B: col N
  const long tok_base =
      ((long)blockIdx.x * (long)(blockDim.x >> 5) + wave) * 16;

  const float* arow = feat + (tok_base + mn) * (long)H;
  const bool   active = (mn < D);
  const float* brow = sW + (long)(active ? mn : D) * HP;  // zero row if !active
  const int    kb = 2 * half;

  v8f c0 = {0.f, 0.f, 0.f, 0.f, 0.f, 0.f, 0.f, 0.f};
  v8f c1 = {0.f, 0.f, 0.f, 0.f, 0.f, 0.f, 0.f, 0.f};
  #pragma unroll 4
  for (int kk = 0; kk < H; kk += 8) {
    v2f a0 = *(const v2f*)(arow + kk + kb);       // global_load_b64 (HBM)
    v2f b0 = *(const v2f*)(brow + kk + kb);       // ds_load_b64 (LDS W panel)
    v2f a1 = *(const v2f*)(arow + kk + 4 + kb);
    v2f b1 = *(const v2f*)(brow + kk + 4 + kb);
    c0 = __builtin_amdgcn_wmma_f32_16x16x4_f32(false, a0, false, b0,
                                               (short)0, c0, false, false);
    c1 = __builtin_amdgcn_wmma_f32_16x16x4_f32(false, a1, false, b1,
                                               (short)0, c1, false, false);
  }

  // C layout: VGPR r -> (M=r, N=lane) for lanes 0-15, (M=8+r) for lanes 16-31.
  if (active) {
    float* p = proj + (tok_base + (long)half * 8) * D + mn;
    #pragma unroll
    for (int r = 0; r < 8; ++r) p[(long)r * D] = c0[r] + c1[r];
  }
}

// ---------------------------------------------------------------------------
// Pass 2: deterministic bin-reduce + bias. One thread per (batch, bin).
// time[b] row staged in LDS (16 KB); every thread scans tokens in a fixed
// order -> bit-deterministic (no float atomics). proj reads hit L2 (256 KB).
//   out[b,m,d]  = sum_{t<len, time==m} proj[t,d] / max(cnt,1) + b_out[d]
//   mask[b,m]   = (no token with raw time == m)
// ---------------------------------------------------------------------------
__global__ __launch_bounds__(256) void br_pool_head_kernel(
    const float* __restrict__ proj, const int* __restrict__ time,
    const int* __restrict__ lengths, const float* __restrict__ b_out,
    float* __restrict__ out, float* __restrict__ mask,
    int T, int TM, int D)
{
  __shared__ int s_time[4096];                // T ints
  const int tid = threadIdx.x;
  const int blocksPerBatch = TM / (int)blockDim.x;   // 512/256 = 2
  const int b = blockIdx.x / blocksPerBatch;
  const int m = (blockIdx.x % blocksPerBatch) * (int)blockDim.x + tid;

  const int* trow = time + (long)b * T;
  for (int i = tid; i < T; i += blockDim.x) s_time[i] = trow[i];
  __syncthreads();

  const int len = lengths[b];
  const float* prow = proj + (long)b * T * D;

  float s0 = 0.f, s1 = 0.f;
  int cv = 0, c2 = 0;
  for (int t = 0; t < T; ++t) {
    if (s_time[t] == m) {                     // raw-time match
      ++c2;
      if (t < len) {                          // non-padded token
        ++cv;
        s0 += prow[(long)t * D + 0];
        s1 += prow[(long)t * D + 1];
      }
    }
  }
  const float dv = (float)(cv > 0 ? cv : 1);
  const long  o  = ((long)b * TM + m) * D;
  out[o + 0] = s0 / dv + b_out[0];
  out[o + 1] = s1 / dv + b_out[1];
  mask[(long)b * TM + m] = (c2 == 0) ? 1.0f : 0.0f;
}

extern "C" void kernel_launch(void* const* d_in, const int* in_sizes, int n_in,
                              void* d_out, int out_size, void* d_ws, size_t ws_size,
                              hipStream_t stream) {
  const float* feat    = (const float*)d_in[0];   // (B,T,H) f32
  const int*   time    = (const int*)  d_in[1];   // (B,T)   i32
  const int*   lengths = (const int*)  d_in[2];   // (B,)    i32
  const float* W       = (const float*)d_in[4];   // (D,H)   f32
  const float* b_out   = (const float*)d_in[5];   // (D,)    f32

  const int  B  = in_sizes[2];
  const long BT = (long)in_sizes[1];              // B*T tokens
  const int  T  = (int)(BT / B);
  const int  H  = (int)((long)in_sizes[0] / BT);
  const int  D  = in_sizes[4] / H;
  const int  TM = out_size / (B * (D + 1));       // out (B,TM,D) + mask (B,TM)

  float* proj = (float*)d_ws;                     // (B*T, D) scratch, 256 KB
  float* out  = (float*)d_out;                    // first B*TM*D floats
  float* mask = out + (long)B * TM * D;           // then B*TM mask floats

  const int blocksA = (int)(BT / 16 / 8);         // 16 tok/wave, 8 waves/block
  br_proj_wmma_kernel<<<blocksA, 256, 0, stream>>>(feat, W, proj, H, D);

  const int blocksB = (B * TM) / 256;
  br_pool_head_kernel<<<blocksB, 256, 0, stream>>>(proj, time, lengths, b_out,
                                                   out, mask, T, TM, D);
}